// RoIAlign_30262339568228
// MI455X (gfx1250) — compile-verified
//
#include <hip/hip_runtime.h>

typedef __attribute__((ext_vector_type(2))) float v2f;
typedef __attribute__((ext_vector_type(8))) float v8f;

#define RA_H      200
#define RA_W      200
#define RA_C      256
#define RA_SCALE  0.0625f
#define RA_POOL   7
#define RA_NBINS  49      // 7*7
#define RA_NSAMP  196     // 49 bins * 4 samples (2x2 grid)

// One block per RoI. 256 threads = 8 waves.
// Phase A: 196 threads compute per-sample tap offsets + bilinear weights -> LDS.
// Phase B: each wave owns 2 tiles of 16 channels; pooling contraction done with
//          v_wmma_f32_16x16x4_f32 (D[16c,16bin] += A[16c,4samp] x B[4samp,16bin]),
//          B is the 0.25 bin-selector (all 4 K-rows identical).
__global__ __launch_bounds__(256) void roi_align_wmma_kernel(
    const float* __restrict__ feat,   // [2,256,200,200]
    const float* __restrict__ rois,   // [nroi,5]
    float* __restrict__ out)          // [nroi,256,7,7]
{
    __shared__ int   s_off[RA_NSAMP];      // y0*W + x0
    __shared__ int   s_dx [RA_NSAMP];      // 0 or 1
    __shared__ int   s_dy [RA_NSAMP];      // 0 or W
    __shared__ float s_w  [RA_NSAMP][4];   // bilinear weights (zeroed if invalid)
    __shared__ int   s_batch;

    const int r   = blockIdx.x;
    const int tid = threadIdx.x;

    if (tid == 0) s_batch = (int)rois[(size_t)r * 5 + 0];

    if (tid < RA_NSAMP) {
        const int bin = tid >> 2;          // 0..48
        const int sub = tid & 3;           // iy*2+ix
        const int ph  = bin / RA_POOL;
        const int pw  = bin % RA_POOL;
        const int iy  = sub >> 1;
        const int ix  = sub & 1;

        const float x1 = rois[(size_t)r * 5 + 1] * RA_SCALE;
        const float y1 = rois[(size_t)r * 5 + 2] * RA_SCALE;
        const float x2 = rois[(size_t)r * 5 + 3] * RA_SCALE;
        const float y2 = rois[(size_t)r * 5 + 4] * RA_SCALE;
        const float roi_w = fmaxf(x2 - x1, 1.0f);
        const float roi_h = fmaxf(y2 - y1, 1.0f);
        const float bin_h = roi_h * (1.0f / RA_POOL);
        const float bin_w = roi_w * (1.0f / RA_POOL);

        float y = y1 + (float)ph * bin_h + ((float)iy + 0.5f) * bin_h * 0.5f;
        float x = x1 + (float)pw * bin_w + ((float)ix + 0.5f) * bin_w * 0.5f;

        const bool valid = (y > -1.0f) && (y < (float)RA_H) &&
                           (x > -1.0f) && (x < (float)RA_W);
        y = fmaxf(y, 0.0f);
        x = fmaxf(x, 0.0f);
        int y0 = min((int)floorf(y), RA_H - 1);
        int x0 = min((int)floorf(x), RA_W - 1);
        const int y1i = min(y0 + 1, RA_H - 1);
        const int x1i = min(x0 + 1, RA_W - 1);
        if (y0 >= RA_H - 1) y = (float)y0;   // snap -> ly = 0
        if (x0 >= RA_W - 1) x = (float)x0;   // snap -> lx = 0
        const float ly = y - (float)y0;
        const float lx = x - (float)x0;
        const float hy = 1.0f - ly;
        const float hx = 1.0f - lx;
        const float vf = valid ? 1.0f : 0.0f;

        s_off[tid]  = y0 * RA_W + x0;
        s_dx[tid]   = x1i - x0;              // 0 or 1
        s_dy[tid]   = (y1i - y0) * RA_W;     // 0 or 200
        s_w[tid][0] = hy * hx * vf;
        s_w[tid][1] = hy * lx * vf;
        s_w[tid][2] = ly * hx * vf;
        s_w[tid][3] = ly * lx * vf;
    }
    __syncthreads();

    const int wave  = tid >> 5;        // 0..7
    const int lane  = tid & 31;
    const int laneM = lane & 15;       // channel-in-tile (A) == column N (B/D)
    const bool hiHalf = (lane >= 16);  // A: K=2,3 ; D: M += 8

    const size_t imgBase = (size_t)s_batch * (size_t)RA_C * RA_H * RA_W;

    for (int ct = 0; ct < 2; ++ct) {
        const int c = wave * 32 + ct * 16 + laneM;            // 0..255
        const float* __restrict__ fc = feat + imgBase + (size_t)c * (RA_H * RA_W);

        for (int jt = 0; jt < 4; ++jt) {                      // bin tiles (49 -> 64)
            v8f acc = {0.f, 0.f, 0.f, 0.f, 0.f, 0.f, 0.f, 0.f};

            for (int t = 0; t < 16; ++t) {                    // K-steps: one bin each
                const int bin    = jt * 16 + t;               // wave-uniform
                const bool pad   = (bin >= RA_NBINS);         // wave-uniform
                const int  sbase = (pad ? 0 : bin * 4) + (hiHalf ? 2 : 0);

                float a0, a1;
                {
                    const int k = sbase;
                    const float* p = fc + s_off[k];
                    a0 = s_w[k][0] * p[0]
                       + s_w[k][1] * p[s_dx[k]]
                       + s_w[k][2] * p[s_dy[k]]
                       + s_w[k][3] * p[s_dy[k] + s_dx[k]];
                }
                {
                    const int k = sbase + 1;
                    const float* p = fc + s_off[k];
                    a1 = s_w[k][0] * p[0]
                       + s_w[k][1] * p[s_dx[k]]
                       + s_w[k][2] * p[s_dy[k]]
                       + s_w[k][3] * p[s_dy[k] + s_dx[k]];
                }
                if (pad) { a0 = 0.0f; a1 = 0.0f; }            // keep A finite for padding

                v2f A = {a0, a1};
                const float bv = (laneM == t) ? 0.25f : 0.0f; // all 4 B rows identical
                v2f B = {bv, bv};

                acc = __builtin_amdgcn_wmma_f32_16x16x4_f32(
                          /*neg_a=*/false, A, /*neg_b=*/false, B,
                          /*c_mod=*/(short)0, acc,
                          /*reuse_a=*/false, /*reuse_b=*/false);
            }

            // D layout: N = lane&15 ; M = vgpr + (lane<16 ? 0 : 8)
            const int binN = jt * 16 + laneM;
            if (binN < RA_NBINS) {
                #pragma unroll
                for (int v = 0; v < 8; ++v) {
                    const int M    = v + (hiHalf ? 8 : 0);
                    const int cOut = wave * 32 + ct * 16 + M;
                    out[((size_t)r * RA_C + cOut) * RA_NBINS + binN] = acc[v];
                }
            }
        }
    }
}

extern "C" void kernel_launch(void* const* d_in, const int* in_sizes, int n_in,
                              void* d_out, int out_size, void* d_ws, size_t ws_size,
                              hipStream_t stream) {
    const float* feat = (const float*)d_in[0];   // [2,256,200,200] f32
    const float* rois = (const float*)d_in[1];   // [nroi,5] f32
    float* out = (float*)d_out;                  // [nroi,256,7,7] f32
    const int num_rois = in_sizes[1] / 5;

    roi_align_wmma_kernel<<<num_rois, 256, 0, stream>>>(feat, rois, out);
}